// TEXTnetOrder2_39737037423319
// MI455X (gfx1250) — compile-verified
//
#include <hip/hip_runtime.h>
#include <math.h>

// fp32 WMMA fragment types (gfx1250, wave32):
//  A/B 16x4 f32 -> 2 VGPRs/lane, C/D 16x16 f32 -> 8 VGPRs/lane
typedef __attribute__((ext_vector_type(2))) float v2f;
typedef __attribute__((ext_vector_type(8))) float v8f;

// Probe-revealed operand type of the async-to-LDS builtin: int4* (generic).
typedef int i4v __attribute__((vector_size(4 * sizeof(int))));

typedef __attribute__((address_space(3))) float lds_f;

#define BM 128          // workgroup M tile
#define BN 128          // workgroup N tile
#define KB 32           // K block staged in LDS
#define LDK (KB + 4)    // 36-dword row stride: 16B-aligned rows for async B128,
                        // banks 36L%64 conflict-free across the wave

// ---- CDNA5 async global->LDS copy (ASYNCcnt path), 16 bytes per lane ----
__device__ __forceinline__ void async_copy16(const float* gsrc, float* ldst) {
#if __has_builtin(__builtin_amdgcn_global_load_async_to_lds_b128)
    __builtin_amdgcn_global_load_async_to_lds_b128((i4v*)gsrc, (i4v*)ldst, 0, 0);
#else
    unsigned loff = (unsigned)(uintptr_t)(lds_f*)ldst;
    asm volatile("global_load_async_to_lds_b128 %0, %1, off"
                 :: "v"(loff), "v"(gsrc) : "memory");
#endif
}

__device__ __forceinline__ void wait_async0() {
#if __has_builtin(__builtin_amdgcn_s_wait_asynccnt)
    __builtin_amdgcn_s_wait_asynccnt(0);
#else
    asm volatile("s_wait_asynccnt 0x0" ::: "memory");
#endif
}

// C[M,N] = act( A[M,K] @ W[N,K]^T + biasN + bias2N + addendMN )
// A is a virtual concatenation of up to 4 row-major segments A0..A3 with
// widths K0..K3. All widths / K's are multiples of 4 for this model.
__global__ __launch_bounds__(256, 2)
void gemm_wmma_f32(const float* __restrict__ A0, const float* __restrict__ A1,
                   const float* __restrict__ A2, const float* __restrict__ A3,
                   int K0, int K1, int K2, int K3,
                   const float* __restrict__ W,
                   const float* __restrict__ biasN,
                   const float* __restrict__ bias2N,
                   const float* __restrict__ addendMN,
                   float* __restrict__ C,
                   int M, int N, int K, int act)
{
    __shared__ float Alds[2][BM][LDK];
    __shared__ float Wlds[2][BN][LDK];

    const int tid  = threadIdx.x;
    const int lane = tid & 31;
    const int half = lane >> 4;      // 0: lanes 0-15, 1: lanes 16-31
    const int lr   = lane & 15;
    const int wid  = tid >> 5;       // 8 waves
    const int wm0  = (wid & 3) * 32; // 4 waves along M, 32 rows each
    const int wn0  = (wid >> 2) * 64;// 2 waves along N, 64 cols each
    const int bm   = blockIdx.y * BM;
    const int bn   = blockIdx.x * BN;

    const int E0 = K0, E1 = K0 + K1, E2 = K0 + K1 + K2;

    v8f acc[2][4];
#pragma unroll
    for (int i = 0; i < 2; ++i)
#pragma unroll
        for (int j = 0; j < 4; ++j)
            acc[i][j] = (v8f){0.f,0.f,0.f,0.f,0.f,0.f,0.f,0.f};

    // Stage one BM/BN x KB tile pair into buffer `buf`.
    // Uniform full-tile condition -> async direct-to-LDS (ASYNCcnt);
    // boundary tiles -> guarded VGPR loads + zero fill (DScnt, waited by barrier).
    auto stage = [&](int buf, int kb) {
        const bool kfull = (kb + KB <= K);
        const bool fullA = kfull && (bm + BM <= M);
        const bool fullW = kfull && (bn + BN <= N);
        // ---- A tile (BM x KB): 1024 float4 groups over 256 threads ----
#pragma unroll
        for (int t = 0; t < (BM * (KB / 4)) / 256; ++t) {
            int g   = t * 256 + tid;
            int row = g >> 3;
            int c4  = (g & 7) << 2;
            int kk  = kb + c4;
            int m   = bm + row;
            if (fullA) {
                const float* p; int ldw, col;
                if      (kk < E0) { p = A0; ldw = K0; col = kk;      }
                else if (kk < E1) { p = A1; ldw = K1; col = kk - E0; }
                else if (kk < E2) { p = A2; ldw = K2; col = kk - E1; }
                else              { p = A3; ldw = K3; col = kk - E2; }
                async_copy16(p + (size_t)m * ldw + col, &Alds[buf][row][c4]);
            } else {
                float4 v; v.x = v.y = v.z = v.w = 0.f;
                if (m < M && kk < K) {
                    const float* p; int ldw, col;
                    if      (kk < E0) { p = A0; ldw = K0; col = kk;      }
                    else if (kk < E1) { p = A1; ldw = K1; col = kk - E0; }
                    else if (kk < E2) { p = A2; ldw = K2; col = kk - E1; }
                    else              { p = A3; ldw = K3; col = kk - E2; }
                    v = *(const float4*)(p + (size_t)m * ldw + col);
                }
                *(float4*)&Alds[buf][row][c4] = v;
            }
        }
        // ---- W tile (BN x KB): B[k][n] = W[n][k] ----
#pragma unroll
        for (int t = 0; t < (BN * (KB / 4)) / 256; ++t) {
            int g   = t * 256 + tid;
            int row = g >> 3;
            int c4  = (g & 7) << 2;
            int kk  = kb + c4;
            int n   = bn + row;
            if (fullW) {
                async_copy16(W + (size_t)n * K + kk, &Wlds[buf][row][c4]);
            } else {
                float4 v; v.x = v.y = v.z = v.w = 0.f;
                if (n < N && kk < K)
                    v = *(const float4*)(W + (size_t)n * K + kk);
                *(float4*)&Wlds[buf][row][c4] = v;
            }
        }
    };

    auto compute = [&](int buf) {
#pragma unroll
        for (int k4 = 0; k4 < KB / 4; ++k4) {
            const int kc = (k4 << 2) + (half << 1); // K pair per half-wave
            v2f a[2], b[4];
#pragma unroll
            for (int i = 0; i < 2; ++i) {
                const float* p = &Alds[buf][wm0 + i * 16 + lr][kc];
                a[i].x = p[0]; a[i].y = p[1];
            }
#pragma unroll
            for (int j = 0; j < 4; ++j) {
                const float* p = &Wlds[buf][wn0 + j * 16 + lr][kc];
                b[j].x = p[0]; b[j].y = p[1];
            }
#pragma unroll
            for (int i = 0; i < 2; ++i)
#pragma unroll
                for (int j = 0; j < 4; ++j)
                    acc[i][j] = __builtin_amdgcn_wmma_f32_16x16x4_f32(
                        false, a[i], false, b[j],
                        (short)0, acc[i][j], false, false);
        }
    };

    // ---- double-buffered main loop: prefetch kb+KB while computing kb ----
    int buf = 0;
    stage(0, 0);
    wait_async0();
    __syncthreads();
    for (int kb = 0; kb < K; kb += KB) {
        const int knext = kb + KB;
        if (knext < K) stage(buf ^ 1, knext);
        compute(buf);
        wait_async0();          // next buffer's async loads (this wave) done
        __syncthreads();        // whole workgroup done writing/reading
        buf ^= 1;
    }

    // ---- epilogue: bias / bias2 / elementwise addend / activation ----
#pragma unroll
    for (int j = 0; j < 4; ++j) {
        int n = bn + wn0 + j * 16 + lr;
        float badd = 0.f;
        if (n < N) {
            if (biasN)  badd += biasN[n];
            if (bias2N) badd += bias2N[n];
        }
#pragma unroll
        for (int i = 0; i < 2; ++i) {
            int mbase = bm + wm0 + i * 16 + half * 8;
#pragma unroll
            for (int v = 0; v < 8; ++v) {
                int m = mbase + v;
                if (m < M && n < N) {
                    float val = acc[i][j][v] + badd;
                    if (addendMN) val += addendMN[(size_t)m * N + n];
                    if      (act == 1) val = fmaxf(val, 0.f);
                    else if (act == 2) val = tanhf(val);
                    C[(size_t)m * N + n] = val;
                }
            }
        }
    }
}

static inline void launch_gemm(hipStream_t s,
    const float* A0, int K0, const float* A1, int K1,
    const float* A2, int K2, const float* A3, int K3,
    const float* W, const float* biasN, const float* bias2N,
    const float* addendMN, float* C, int M, int N, int K, int act)
{
    dim3 grid((N + BN - 1) / BN, (M + BM - 1) / BM);
    gemm_wmma_f32<<<grid, dim3(256), 0, s>>>(A0, A1, A2, A3, K0, K1, K2, K3,
                                             W, biasN, bias2N, addendMN,
                                             C, M, N, K, act);
}

extern "C" void kernel_launch(void* const* d_in, const int* in_sizes, int n_in,
                              void* d_out, int out_size, void* d_ws, size_t ws_size,
                              hipStream_t stream) {
    (void)in_sizes; (void)n_in; (void)out_size; (void)ws_size;
    const int B = 4096, I = 1024, H = 2048, O = 256, MD = 100;

    const float* input   = (const float*)d_in[0];
    const float* hidden  = (const float*)d_in[1];
    const float* prevh   = (const float*)d_in[2];
    const float* cell    = (const float*)d_in[3];
    const float* W_ch2   = (const float*)d_in[4];
    const float* b_ch2   = (const float*)d_in[5];
    const float* W_cm2   = (const float*)d_in[6];
    const float* b_cm2   = (const float*)d_in[7];
    const float* W_mo    = (const float*)d_in[8];
    const float* b_mo    = (const float*)d_in[9];
    const float* W_ci    = (const float*)d_in[10];
    const float* b_ci    = (const float*)d_in[11];
    const float* W_ch    = (const float*)d_in[12];
    const float* b_ch    = (const float*)d_in[13];
    const float* W_fc    = (const float*)d_in[14];
    const float* b_fc    = (const float*)d_in[15];
    const float* biasVec = (const float*)d_in[16];
    // d_in[17] = k (static 1)

    float* out       = (float*)d_out;                     // [B, O]
    float* new_hid   = out + (size_t)B * O;               // [B, H]
    float* new_cell  = new_hid + (size_t)B * H;           // [B, H]
    float* new_prevh = new_cell + (size_t)B * H;          // [B, H] (= h_proj)

    float* mid = (float*)d_ws;                            // [B, 100]
    float* lc  = mid + (size_t)B * MD;                    // [B, H]

    // 1) new_hidden = concat @ W_ch2^T + b_ch2
    launch_gemm(stream, input, I, hidden, H, prevh, H, cell, H,
                W_ch2, b_ch2, nullptr, nullptr, new_hid, B, H, I + 3 * H, 0);
    // 2) mid = relu(concat @ W_cm2^T + b_cm2)
    launch_gemm(stream, input, I, hidden, H, prevh, H, cell, H,
                W_cm2, b_cm2, nullptr, nullptr, mid, B, MD, I + 3 * H, 1);
    // 3) out = mid @ W_mo^T + b_mo
    launch_gemm(stream, mid, MD, nullptr, 0, nullptr, 0, nullptr, 0,
                W_mo, b_mo, nullptr, nullptr, out, B, O, MD, 0);
    // 4) h_proj = new_hidden @ W_ch^T + b_ch   -> new_prev_hidden output slot
    launch_gemm(stream, new_hid, H, nullptr, 0, nullptr, 0, nullptr, 0,
                W_ch, b_ch, nullptr, nullptr, new_prevh, B, H, H, 0);
    // 5) lc = input @ W_ci^T + b_ci + h_proj + bias
    launch_gemm(stream, input, I, nullptr, 0, nullptr, 0, nullptr, 0,
                W_ci, b_ci, biasVec, new_prevh, lc, B, H, I, 0);
    // 6) new_cell = tanh(lc @ W_fc^T + b_fc)
    launch_gemm(stream, lc, H, nullptr, 0, nullptr, 0, nullptr, 0,
                W_fc, b_fc, nullptr, nullptr, new_cell, B, H, H, 2);
}